// Net_1786706395263
// MI455X (gfx1250) — compile-verified
//
#include <hip/hip_runtime.h>
#include <math.h>

#define DF 128
#define DH 64

typedef __attribute__((ext_vector_type(2))) float v2f;
typedef __attribute__((ext_vector_type(8))) float v8f;

// ---------------------------------------------------------------------------
// 1) init: deg=1 (self-loop weight), acc1[N*64]=0, a2[N]=0
// ---------------------------------------------------------------------------
__global__ __launch_bounds__(256) void k_init(float* __restrict__ deg,
                                              float* __restrict__ acc,
                                              float* __restrict__ a2, int N) {
    int i = blockIdx.x * blockDim.x + threadIdx.x;
    int total = 66 * N;
    if (i >= total) return;
    if (i < N)            deg[i]          = 1.0f;   // self-loop contributes w=1
    else if (i < 65 * N)  acc[i - N]      = 0.0f;   // 64*N layer-1 accumulator
    else                  a2[i - 65 * N]  = 0.0f;   // layer-2 accumulator
}

// ---------------------------------------------------------------------------
// 2) degree: deg[dst[e]] += w[e]
// ---------------------------------------------------------------------------
__global__ __launch_bounds__(256) void k_deg(const int* __restrict__ dst,
                                             const float* __restrict__ w,
                                             float* __restrict__ deg, int E) {
    int e = blockIdx.x * blockDim.x + threadIdx.x;
    if (e < E) atomicAdd(&deg[dst[e]], w[e]);
}

// ---------------------------------------------------------------------------
// 3) dinv = deg>0 ? rsqrt(deg) : 0   (in place)
// ---------------------------------------------------------------------------
__global__ __launch_bounds__(256) void k_dinv(float* __restrict__ deg, int N) {
    int i = blockIdx.x * blockDim.x + threadIdx.x;
    if (i < N) {
        float d = deg[i];
        deg[i] = (d > 0.0f) ? rsqrtf(fmaxf(d, 1e-30f)) : 0.0f;
    }
}

// ---------------------------------------------------------------------------
// 4) h = x @ W1   via V_WMMA_F32_16X16X4_F32, one 16x16 tile per wave.
//    Block = 128 threads (4 waves) -> one 16-row strip, all 64 columns.
//    A 16x4 f32 layout (ISA 7.12.2): lanes 0-15: K=0(V0),K=1(V1);
//                                    lanes 16-31: K=2(V0),K=3(V1).
//    B 4x16: row-striped across lanes per VGPR (group0/group1 = K, K+2).
//    C/D 16x16: VGPR r -> M=r (lanes 0-15) / M=r+8 (lanes 16-31).
// ---------------------------------------------------------------------------
__global__ __launch_bounds__(128) void k_gemm(const float* __restrict__ x,
                                              const float* __restrict__ W1,
                                              float* __restrict__ h, int N) {
    const int lane   = threadIdx.x & 31;
    const int m      = lane & 15;
    const int half   = lane >> 4;            // 0: lanes 0-15, 1: lanes 16-31
    const int col0   = (threadIdx.x >> 5) * 16;  // wave -> column tile
    const int row0   = blockIdx.x * 16;

    int arow = row0 + m;
    if (arow >= N) arow = N - 1;             // N is a multiple of 16; safe clamp
    const float* __restrict__ xr = x + (size_t)arow * DF;

    v8f c = {};
#pragma unroll
    for (int k = 0; k < DF; k += 4) {
        v2f a, b;
        a.x = xr[k + 2 * half];
        a.y = xr[k + 2 * half + 1];
        b.x = W1[(size_t)(k + 2 * half)     * DH + col0 + m];
        b.y = W1[(size_t)(k + 2 * half + 1) * DH + col0 + m];
        // (neg_a, A, neg_b, B, c_mod, C, reuse_a, reuse_b)
        c = __builtin_amdgcn_wmma_f32_16x16x4_f32(false, a, false, b,
                                                  (short)0, c, false, false);
    }
#pragma unroll
    for (int r = 0; r < 8; ++r) {
        int orow = row0 + r + 8 * half;
        if (orow >= N) orow = N - 1;
        h[(size_t)orow * DH + col0 + m] = c[r];
    }
}

// ---------------------------------------------------------------------------
// 5) layer-1 message pass: one wave32 per (edge or self-loop).
//    Each lane handles 2 of the 64 features: coalesced 256B read of h[src],
//    coalesced 256B of f32 atomic adds at acc[dst].
// ---------------------------------------------------------------------------
__global__ __launch_bounds__(256) void k_agg1(const int* __restrict__ src,
                                              const int* __restrict__ dst,
                                              const float* __restrict__ w,
                                              const float* __restrict__ dinv,
                                              const float* __restrict__ h,
                                              float* __restrict__ acc,
                                              int E, int N) {
    int wid  = (blockIdx.x * blockDim.x + threadIdx.x) >> 5;
    int lane = threadIdx.x & 31;
    if (wid >= E + N) return;
    int s, d; float we;
    if (wid < E) { s = src[wid]; d = dst[wid]; we = w[wid]; }
    else         { s = d = wid - E; we = 1.0f; }            // self loop
    float nm = dinv[s] * we * dinv[d];
    const float2 hv = *(const float2*)(h + (size_t)s * DH + 2 * lane);
    float* ap = acc + (size_t)d * DH + 2 * lane;
    atomicAdd(ap,     nm * hv.x);
    atomicAdd(ap + 1, nm * hv.y);
}

// ---------------------------------------------------------------------------
// 6) per-node: h2[n] = sum_k elu(acc[n,k]+b1[k]) * W2[k]   (wave32 per node)
// ---------------------------------------------------------------------------
__global__ __launch_bounds__(256) void k_node(const float* __restrict__ acc,
                                              const float* __restrict__ b1,
                                              const float* __restrict__ W2,
                                              float* __restrict__ h2, int N) {
    int wid  = (blockIdx.x * blockDim.x + threadIdx.x) >> 5;
    int lane = threadIdx.x & 31;
    if (wid >= N) return;
    const float2 v  = *(const float2*)(acc + (size_t)wid * DH + 2 * lane);
    const float2 bb = *(const float2*)(b1 + 2 * lane);
    const float2 ww = *(const float2*)(W2 + 2 * lane);
    float a0 = v.x + bb.x;  a0 = (a0 > 0.0f) ? a0 : (expf(a0) - 1.0f);  // ELU(1)
    float a1 = v.y + bb.y;  a1 = (a1 > 0.0f) ? a1 : (expf(a1) - 1.0f);
    float p = a0 * ww.x + a1 * ww.y;
#pragma unroll
    for (int off = 16; off > 0; off >>= 1)
        p += __shfl_xor(p, off, 32);                       // wave32 reduction
    if (lane == 0) h2[wid] = p;
}

// ---------------------------------------------------------------------------
// 7) layer-2 message pass (scalar): a2[dst] += norm * h2[src]
// ---------------------------------------------------------------------------
__global__ __launch_bounds__(256) void k_agg2(const int* __restrict__ src,
                                              const int* __restrict__ dst,
                                              const float* __restrict__ w,
                                              const float* __restrict__ dinv,
                                              const float* __restrict__ h2,
                                              float* __restrict__ a2,
                                              int E, int N) {
    int e = blockIdx.x * blockDim.x + threadIdx.x;
    if (e >= E + N) return;
    int s, d; float we;
    if (e < E) { s = src[e]; d = dst[e]; we = w[e]; }
    else       { s = d = e - E; we = 1.0f; }
    float nm = dinv[s] * we * dinv[d];
    atomicAdd(&a2[d], nm * h2[s]);
}

// ---------------------------------------------------------------------------
// 8) out = sigmoid(a2 + b2)
// ---------------------------------------------------------------------------
__global__ __launch_bounds__(256) void k_out(const float* __restrict__ a2,
                                             const float* __restrict__ b2,
                                             float* __restrict__ out, int N) {
    int i = blockIdx.x * blockDim.x + threadIdx.x;
    if (i < N) {
        float z = a2[i] + b2[0];
        out[i] = 1.0f / (1.0f + expf(-z));
    }
}

extern "C" void kernel_launch(void* const* d_in, const int* in_sizes, int n_in,
                              void* d_out, int out_size, void* d_ws, size_t ws_size,
                              hipStream_t stream) {
    const float* x   = (const float*)d_in[0];
    const int*   ei  = (const int*)  d_in[1];   // [2,E] flat: src then dst
    const float* ew  = (const float*)d_in[2];
    const float* W1  = (const float*)d_in[3];
    const float* b1  = (const float*)d_in[4];
    const float* W2  = (const float*)d_in[5];
    const float* b2  = (const float*)d_in[6];
    float* out = (float*)d_out;

    const int N = in_sizes[0] / DF;
    const int E = in_sizes[2];
    const int* src = ei;
    const int* dst = ei + E;

    // workspace layout (floats): deg/dinv[N] | h[64N] | acc[64N] | h2[N] | a2[N]
    float* ws  = (float*)d_ws;
    float* deg = ws;
    float* h   = deg + (size_t)N;
    float* acc = h   + (size_t)N * DH;
    float* h2  = acc + (size_t)N * DH;
    float* a2  = h2  + (size_t)N;

    const int B = 256;
    // 1) init (must run every call: graph replays do not re-zero anything)
    k_init<<<(66 * N + B - 1) / B, B, 0, stream>>>(deg, acc, a2, N);
    // 2) degrees
    k_deg<<<(E + B - 1) / B, B, 0, stream>>>(dst, ew, deg, E);
    // 3) dinv
    k_dinv<<<(N + B - 1) / B, B, 0, stream>>>(deg, N);
    // 4) h = x @ W1 (WMMA f32), one 16-row strip per block
    k_gemm<<<(N + 15) / 16, 128, 0, stream>>>(x, W1, h, N);
    // 5) layer-1 scatter: one wave per edge (+self loops)
    k_agg1<<<((E + N) + 7) / 8, B, 0, stream>>>(src, dst, ew, deg, h, acc, E, N);
    // 6) fused bias+ELU+dot(W2): one wave per node
    k_node<<<(N + 7) / 8, B, 0, stream>>>(acc, b1, W2, h2, N);
    // 7) layer-2 scalar scatter
    k_agg2<<<((E + N) + B - 1) / B, B, 0, stream>>>(src, dst, ew, deg, h2, a2, E, N);
    // 8) sigmoid epilogue
    k_out<<<(N + B - 1) / B, B, 0, stream>>>(a2, b2, out, N);
}